// SunShapeBlockCodec_77421080478375
// MI455X (gfx1250) — compile-verified
//
#include <hip/hip_runtime.h>
#include <hip/hip_bf16.h>

typedef __attribute__((ext_vector_type(2))) float v2f;
typedef __attribute__((ext_vector_type(8))) float v8f;

#define NROWS     262144
#define HEAD_DIM  128
#define NBLOCKS   16
#define NCENT     256
#define BDIM      8
#define CHUNKS_PER_WG 8
#define ROWS_PER_WG   (16 * CHUNKS_PER_WG)   // 128 rows per workgroup

// One workgroup = 16 waves (512 threads). Wave w handles VQ block w for the
// workgroup's 128-row tile. Centroids live in VGPRs as WMMA B-operands.
// Scores are ||x-c||^2 + 1 (> 0), computed entirely in the matrix pipe via a
// third "augmentation" WMMA per tile injecting ||c||^2 + ||x||^2 + 1, so the
// argmin is a 2-op unsigned-key min per accumulator register.
// Register-pressure tricks (round 2/3 spilled at the 256-VGPR edge):
//  - one aug-B pair serves TWO tiles: ||c||^2 of tile 2j on lanes 0-15 (K0
//    row) and of tile 2j+1 on lanes 16-31 (K2 row); constant aug-A selectors
//    pick K0/K1 (even tile) or K2/K3 (odd tile).
//  - tiles processed 4 at a time with sched_barrier to cap live accumulators.
__global__ __launch_bounds__(512)
void SunShapeBlockCodec_vq_kernel(const float* __restrict__ x,
                                  const float* __restrict__ cent,
                                  const int*   __restrict__ perm,
                                  float*       __restrict__ out) {
  const int tid    = threadIdx.x;
  const int wave   = tid >> 5;        // block index b = wave (0..15)
  const int lane   = tid & 31;
  const int lm     = lane & 15;       // position within 16-lane half
  const int hiHalf = lane >> 4;       // 0: lanes 0-15, 1: lanes 16-31
  const int b      = wave;
  const int rowbase0 = blockIdx.x * ROWS_PER_WG;

  // Forward-permuted column indices for this block.
  // K-split: lanes 0-15 carry K={0,1} ({4,5} hi), lanes 16-31 K={2,3} ({6,7}).
  const int kbase = hiHalf ? 2 : 0;
  const int pb    = b * BDIM;
  const int pc0 = perm[pb + kbase + 0];
  const int pc1 = perm[pb + kbase + 1];
  const int pc2 = perm[pb + kbase + 4];
  const int pc3 = perm[pb + kbase + 5];
  const int pOut = perm[pb + (lane & 7)];   // output scatter column (lm<8 lanes)

  // Aug-A K0/K2 selectors: even tiles take B rows K0/K1 (lanes 0-15 side),
  // odd tiles take B rows K2/K3 (lanes 16-31 side).
  const float aEx = hiHalf ? 0.0f : 1.0f;
  const float aOx = hiHalf ? 1.0f : 0.0f;

  // ---- Load B operands (centroid columns), shared aug-B pairs, indices -----
  const unsigned cbOff = (unsigned)b * (NCENT * BDIM);
  const float*   cb    = cent + cbOff;
  v2f bLo[16], bHi[16];
  v2f bNP[8];          // pair j: .x = ||c||^2 of tile 2j (lanes<16) /
                       //               tile 2j+1 (lanes>=16); .y = 1.0
  int vIdxE[8];        // packed centroid index for even tile 2j
#pragma unroll
  for (int j = 0; j < 8; ++j) {
    float full[2];
#pragma unroll
    for (int s = 0; s < 2; ++s) {
      const int t = 2 * j + s;
      const unsigned co = (unsigned)(t * 16 + lm) * BDIM + (unsigned)kbase;
      v2f lo = *(const v2f*)(cb + co);       // components kbase, kbase+1
      v2f hi = *(const v2f*)(cb + co + 4);   // components kbase+4, kbase+5
      bLo[t] = lo;
      bHi[t] = hi;
      // ||c||^2: this lane has 4 of 8 components; partner half has the rest.
      float pN = lo.x * lo.x + lo.y * lo.y + hi.x * hi.x + hi.y * hi.y;
      float qN = __int_as_float(
          __builtin_amdgcn_ds_swizzle(__float_as_int(pN), 0x401f)); // SWAPX16
      full[s] = pN + qN;
    }
    v2f bn;
    bn.x = hiHalf ? full[1] : full[0];
    bn.y = 1.0f;
    bNP[j]   = bn;
    vIdxE[j] = (2 * j) * 16 + lm;   // odd tile index = this | 16 (bit 4 free)
  }

  // ---- Row-chunk loop ------------------------------------------------------
#pragma unroll 1
  for (int c = 0; c < CHUNKS_PER_WG; ++c) {
    const int rowbase = rowbase0 + c * 16;
    const unsigned rowOff = (unsigned)(rowbase + lm) * HEAD_DIM;

    if (c + 1 < CHUNKS_PER_WG) {
      __builtin_prefetch(x + (rowOff + 16u * HEAD_DIM), 0, 1); // global_prefetch_b8
    }

    const float r0 = x[rowOff + (unsigned)pc0];
    const float r1 = x[rowOff + (unsigned)pc1];
    const float r2 = x[rowOff + (unsigned)pc2];
    const float r3 = x[rowOff + (unsigned)pc3];

    // ||x||^2 + 1 for this row (per-row constant; +1 keeps scores positive so
    // float bits of scores order correctly as unsigned ints).
    float p = r0 * r0 + r1 * r1 + r2 * r2 + r3 * r3;
    float q = __int_as_float(
        __builtin_amdgcn_ds_swizzle(__float_as_int(p), 0x401f)); // SWAPX16
    const float xn1 = p + q + 1.0f;

    v2f aLo, aHi, aEv, aOd;
    aLo.x = -2.0f * r0; aLo.y = -2.0f * r1;
    aHi.x = -2.0f * r2; aHi.y = -2.0f * r3;
    aEv.x = aEx;  aEv.y = xn1 * aEx;   // even tiles: K0 = 1, K1 = ||x||^2+1
    aOd.x = aOx;  aOd.y = xn1 * aOx;   // odd tiles:  K2 = 1, K3 = ||x||^2+1

    unsigned mink[8];
#pragma unroll
    for (int r = 0; r < 8; ++r) mink[r] = 0xFFFFFFFFu;

    // 16 N-tiles cover 256 centroids: D = -2 x.c + ||c||^2 + ||x||^2 + 1.
    // Groups of 4 tiles (2 pairs); sched_barrier caps live accumulators.
#pragma unroll
    for (int g = 0; g < 4; ++g) {
#pragma unroll
      for (int jj = 0; jj < 2; ++jj) {
        const int j = g * 2 + jj;

        { // even tile 2j
          v8f acc = {};
          acc = __builtin_amdgcn_wmma_f32_16x16x4_f32(
              false, aLo, false, bLo[2 * j], (short)0, acc, false, false);
          acc = __builtin_amdgcn_wmma_f32_16x16x4_f32(
              false, aHi, false, bHi[2 * j], (short)0, acc, false, false);
          acc = __builtin_amdgcn_wmma_f32_16x16x4_f32(
              false, aEv, false, bNP[j],     (short)0, acc, false, false);
          const unsigned idxE = (unsigned)vIdxE[j];
#pragma unroll
          for (int r = 0; r < 8; ++r) {
            unsigned key = (__float_as_uint(acc[r]) & 0xFFFFFF00u) | idxE;
            mink[r] = key < mink[r] ? key : mink[r];
          }
        }

        { // odd tile 2j+1
          v8f acc = {};
          acc = __builtin_amdgcn_wmma_f32_16x16x4_f32(
              false, aLo, false, bLo[2 * j + 1], (short)0, acc, false, false);
          acc = __builtin_amdgcn_wmma_f32_16x16x4_f32(
              false, aHi, false, bHi[2 * j + 1], (short)0, acc, false, false);
          acc = __builtin_amdgcn_wmma_f32_16x16x4_f32(
              false, aOd, false, bNP[j],         (short)0, acc, false, false);
          const unsigned idxO = (unsigned)vIdxE[j] | 16u;
#pragma unroll
          for (int r = 0; r < 8; ++r) {
            unsigned key = (__float_as_uint(acc[r]) & 0xFFFFFF00u) | idxO;
            mink[r] = key < mink[r] ? key : mink[r];
          }
        }
      }
      __builtin_amdgcn_sched_barrier(0);
    }

    // Cross-lane argmin over the 16 lanes of each half (rows r / r+8).
#pragma unroll
    for (int r = 0; r < 8; ++r) {
      unsigned key = mink[r], o;
      o = (unsigned)__builtin_amdgcn_ds_swizzle((int)key, 0x041f); // xor 1
      key = o < key ? o : key;
      o = (unsigned)__builtin_amdgcn_ds_swizzle((int)key, 0x081f); // xor 2
      key = o < key ? o : key;
      o = (unsigned)__builtin_amdgcn_ds_swizzle((int)key, 0x101f); // xor 4
      key = o < key ? o : key;
      o = (unsigned)__builtin_amdgcn_ds_swizzle((int)key, 0x201f); // xor 8
      key = o < key ? o : key;

      const unsigned idx = key & 255u;
      if ((lane & 15) < 8) {
        // lanes 0-7 -> row rowbase+r, lanes 16-23 -> row rowbase+r+8;
        // lane e writes codeword element e at forward-permuted column.
        const unsigned row = (unsigned)(rowbase + r) + (unsigned)hiHalf * 8u;
        const float val = cb[idx * BDIM + (unsigned)(lane & 7)];
        out[row * (unsigned)HEAD_DIM + (unsigned)pOut] = val;
      }
    }
  }
}

extern "C" void kernel_launch(void* const* d_in, const int* in_sizes, int n_in,
                              void* d_out, int out_size, void* d_ws, size_t ws_size,
                              hipStream_t stream) {
  (void)in_sizes; (void)n_in; (void)out_size; (void)d_ws; (void)ws_size;
  const float* x    = (const float*)d_in[0];   // (N, 128) f32
  const float* cent = (const float*)d_in[1];   // (16, 256, 8) f32
  const int*   perm = (const int*)d_in[2];     // (128,) i32
  float* out = (float*)d_out;                  // (N, 128) f32

  const int grid = NROWS / ROWS_PER_WG;        // 2048 workgroups
  SunShapeBlockCodec_vq_kernel<<<grid, 512, 0, stream>>>(x, cent, perm, out);
}